// KSCD_IF_13572096655713
// MI455X (gfx1250) — compile-verified
//
#include <hip/hip_runtime.h>
#include <hip/hip_bf16.h>
#include <math.h>

// Problem constants (from reference): B=2048, K=128, L=64
#define NB 2048
#define NK 128
#define NL 64
#define W_STRIDE (NK + NL)   // 192: W1/W2 row stride

typedef __attribute__((ext_vector_type(2))) float v2f;
typedef __attribute__((ext_vector_type(8))) float v8f;

__device__ __forceinline__ float fast_rcp(float x) {
#if __has_builtin(__builtin_amdgcn_rcpf)
    return __builtin_amdgcn_rcpf(x);       // v_rcp_f32
#else
    return 1.0f / x;
#endif
}

__device__ __forceinline__ float fast_exp2(float x) {
#if __has_builtin(__builtin_amdgcn_exp2f)
    return __builtin_amdgcn_exp2f(x);      // v_exp_f32
#else
    return exp2f(x);
#endif
}

__device__ __forceinline__ float fast_sigmoid(float x) {
    // 1 / (1 + 2^(-x*log2 e)); exp2->inf gives rcp->0, correct saturation
    return fast_rcp(1.0f + fast_exp2(-1.44269504088896341f * x));
}

__device__ __forceinline__ float fast_tanh(float x) {
#if __has_builtin(__builtin_amdgcn_tanhf)
    return __builtin_amdgcn_tanhf(x);      // v_tanh_f32 (gfx1250 TRANS op)
#else
    // tanh(x) = 1 - 2/(exp(2x)+1); exp2->inf => rcp->0 => +1, exp2->0 => -1
    float e = fast_exp2(2.88539008177792681f * x);   // 2*log2(e)*x
    return 1.0f - 2.0f * fast_rcp(e + 1.0f);
#endif
}

// ---------------------------------------------------------------------------
// K12: fused  S = sigmoid(X @ knowledge^T)  ;  A = S @ |Ws|^T
// One wave (32 threads) per block handles a 16-row strip of B.
// blockIdx.x: b-tile (B/16).  blockIdx.y: 0 -> (student,W1,A1), 1 -> (diff,W2,A2)
// Uses V_WMMA_F32_16X16X4_F32, full fp32 precision.
// ---------------------------------------------------------------------------
__global__ void __launch_bounds__(32)
kscd_fused_gemm(const float* __restrict__ student,
                const float* __restrict__ diffts,
                const float* __restrict__ knowledge,
                const float* __restrict__ W1,
                const float* __restrict__ W2,
                float* __restrict__ A1,
                float* __restrict__ A2) {
    __shared__ float S[16 * NK];   // 8 KB sigmoid intermediate (16 x 128)

    const int lane = threadIdx.x;          // 0..31
    const int l16  = lane & 15;
    const int half = lane >> 4;            // 0 or 1
    const int m0   = blockIdx.x * 16;

    const float* X    = (blockIdx.y == 0) ? student : diffts;
    const float* W    = (blockIdx.y == 0) ? W1 : W2;
    float*       Aout = (blockIdx.y == 0) ? A1 : A2;

    // -------- GEMM1: S[m, n] = sigmoid( sum_k X[m0+m, k] * knowledge[n, k] )
    // A frag (16x4 f32): lane l<16 -> rows, vgpr j holds K = 4*k0 + 2*half + j
    const float* abase = X + (m0 + l16) * NL + 2 * half;
    for (int nt = 0; nt < 8; ++nt) {
        const float* bbase = knowledge + (nt * 16 + l16) * NL + 2 * half;
        v8f acc = {};
#pragma unroll
        for (int k0 = 0; k0 < NL / 4; ++k0) {
            v2f av = *(const v2f*)(abase + 4 * k0);
            v2f bv = *(const v2f*)(bbase + 4 * k0);
            acc = __builtin_amdgcn_wmma_f32_16x16x4_f32(
                false, av, false, bv, (short)0, acc, false, false);
        }
        // D layout: vgpr r -> row (r + 8*half), col l16 within tile
#pragma unroll
        for (int r = 0; r < 8; ++r)
            S[(r + 8 * half) * NK + nt * 16 + l16] = fast_sigmoid(acc[r]);
    }
    __syncthreads();

    // -------- GEMM2: A[m0+m, n] = sum_k S[m, k] * |W[n, k]|   (k over 0..127)
    for (int nt = 0; nt < 8; ++nt) {
        const float* wrow = W + (nt * 16 + l16) * W_STRIDE + 2 * half;
        v8f acc = {};
#pragma unroll
        for (int k0 = 0; k0 < NK / 4; ++k0) {
            v2f av = *(const v2f*)&S[l16 * NK + 4 * k0 + 2 * half];
            v2f wv = *(const v2f*)(wrow + 4 * k0);
            v2f bv;
            bv.x = fabsf(wv.x);   // _pos(w) == |w|
            bv.y = fabsf(wv.y);
            acc = __builtin_amdgcn_wmma_f32_16x16x4_f32(
                false, av, false, bv, (short)0, acc, false, false);
        }
#pragma unroll
        for (int r = 0; r < 8; ++r)
            Aout[(m0 + r + 8 * half) * NK + nt * 16 + l16] = acc[r];
    }
}

// ---------------------------------------------------------------------------
// K3: Ct[c, i] = sum_l knowledge[i, l] * |W[c, 128 + l]|   (stored TRANSPOSED)
// One wave per 16x16 tile.  blockIdx = (i-tile, c-tile, which-W)
// ---------------------------------------------------------------------------
__global__ void __launch_bounds__(32)
kscd_c_gemm(const float* __restrict__ knowledge,
            const float* __restrict__ W1,
            const float* __restrict__ W2,
            float* __restrict__ C1t,
            float* __restrict__ C2t) {
    const int lane = threadIdx.x;
    const int l16  = lane & 15;
    const int half = lane >> 4;
    const int i0   = blockIdx.x * 16;
    const int c0   = blockIdx.y * 16;

    const float* W  = (blockIdx.z == 0) ? W1 : W2;
    float*       Ct = (blockIdx.z == 0) ? C1t : C2t;

    const float* abase = knowledge + (i0 + l16) * NL + 2 * half;
    const float* wrow  = W + (c0 + l16) * W_STRIDE + NK + 2 * half;
    v8f acc = {};
#pragma unroll
    for (int k0 = 0; k0 < NL / 4; ++k0) {
        v2f av = *(const v2f*)(abase + 4 * k0);
        v2f wv = *(const v2f*)(wrow + 4 * k0);
        v2f bv;
        bv.x = fabsf(wv.x);
        bv.y = fabsf(wv.y);
        acc = __builtin_amdgcn_wmma_f32_16x16x4_f32(
            false, av, false, bv, (short)0, acc, false, false);
    }
    // D element: row i = i0 + r + 8*half, col c = c0 + l16 ; store transposed
#pragma unroll
    for (int r = 0; r < 8; ++r)
        Ct[(c0 + l16) * NK + i0 + r + 8 * half] = acc[r];
}

// ---------------------------------------------------------------------------
// K4: fused tanh-contraction + sigmoid + masked mean.
// One block per b (128 threads, thread = i).  C1t/C2t transposed so the
// c-loop loads are lane-coalesced.  ~67M v_exp/v_tanh TRANS ops total.
// ---------------------------------------------------------------------------
__global__ void __launch_bounds__(128)
kscd_main(const float* __restrict__ A1, const float* __restrict__ A2,
          const float* __restrict__ C1t, const float* __restrict__ C2t,
          const float* __restrict__ W3, const float* __restrict__ b3,
          const float* __restrict__ qmask, float* __restrict__ out) {
    __shared__ float a1s[NK];
    __shared__ float a2s[NK];
    __shared__ float red[2 * NK];

    const int i = threadIdx.x;       // exercise index 0..127
    const int b = blockIdx.x;

    a1s[i] = A1[b * NK + i];
    a2s[i] = A2[b * NK + i];
    __syncthreads();

    float acc = 0.0f;
#pragma unroll 4
    for (int c = 0; c < NK; ++c) {
        float w3c = fabsf(W3[c]);                       // _pos == abs
        float t1 = fast_tanh(a1s[c] + C1t[c * NK + i]); // coalesced over lanes
        float t2 = fast_tanh(a2s[c] + C2t[c * NK + i]);
        acc += (t1 - t2) * w3c;
    }
    float o = fast_sigmoid(acc + b3[0]);
    float q = qmask[b * NK + i];

    red[i]      = o * q;
    red[NK + i] = q;
    __syncthreads();
#pragma unroll
    for (int s = 64; s > 0; s >>= 1) {
        if (i < s) {
            red[i]      += red[i + s];
            red[NK + i] += red[NK + i + s];
        }
        __syncthreads();
    }
    if (i == 0) out[b] = red[0] * fast_rcp(red[NK]);
}

// ---------------------------------------------------------------------------
extern "C" void kernel_launch(void* const* d_in, const int* in_sizes, int n_in,
                              void* d_out, int out_size, void* d_ws, size_t ws_size,
                              hipStream_t stream) {
    const float* student   = (const float*)d_in[0];  // (B, L)
    const float* diffts    = (const float*)d_in[1];  // (B, L)
    // d_in[2] = disc_ts : unused by the reference
    const float* knowledge = (const float*)d_in[3];  // (K, L)
    const float* qmask     = (const float*)d_in[4];  // (B, K)
    const float* W1        = (const float*)d_in[5];  // (K, K+L)
    const float* W2        = (const float*)d_in[6];  // (K, K+L)
    const float* W3        = (const float*)d_in[7];  // (1, K)
    const float* b3        = (const float*)d_in[8];  // (1,)
    float* out = (float*)d_out;

    float* ws  = (float*)d_ws;
    float* A1  = ws;                       // B*K floats
    float* A2  = A1 + (size_t)NB * NK;     // B*K floats
    float* C1t = A2 + (size_t)NB * NK;     // K*K floats (transposed)
    float* C2t = C1t + (size_t)NK * NK;    // K*K floats (transposed)
    // total: 2*B*K + 2*K*K floats ~= 2.2 MB of d_ws

    kscd_fused_gemm<<<dim3(NB / 16, 2), 32, 0, stream>>>(
        student, diffts, knowledge, W1, W2, A1, A2);
    kscd_c_gemm<<<dim3(NK / 16, NK / 16, 2), 32, 0, stream>>>(
        knowledge, W1, W2, C1t, C2t);
    kscd_main<<<NB, NK, 0, stream>>>(A1, A2, C1t, C2t, W3, b3, qmask, out);
}